// GlimpseClouds_1632087572806
// MI455X (gfx1250) — compile-verified
//
#include <hip/hip_runtime.h>
#include <hip/hip_bf16.h>
#include <math.h>

typedef float v2f __attribute__((ext_vector_type(2)));
typedef float v8f __attribute__((ext_vector_type(8)));

#define B_ 64
#define T_ 16
#define G_ 3
#define D_ 2048
#define H_ 512
#define C_ 60
#define WK_ 3
#define TH3_ 1536
#define NR_ 48               // T_*G_ rows per batch element
#define TEMP_INV 2.0f        // 1/TEMP
#define CLAMP_MAX_ 100000.0f

// ---------------------------------------------------------------------------
// Generic fp32 WMMA GEMM:  C[M x N] = A[M x K](row) * W[N x K](row)^T
// block = 256 threads (8 wave32), tile 64(M) x 128(N), K staged 16 via LDS.
// Waves arranged 4(M) x 2(N); each wave owns 4 16x16 f32 accumulators and
// issues v_wmma_f32_16x16x4_f32 on K-slices of 4.
// Rows are clamped on load (harmless dup), stores are bounds-guarded, so any
// M,N work as long as K % 16 == 0 (all our K are 2048 or 512).
// ---------------------------------------------------------------------------
__global__ __launch_bounds__(256)
void gemm_at_wt(const float* __restrict__ A, const float* __restrict__ W,
                float* __restrict__ Cp,
                int M, int N, int K, int lda, int ldw, int ldc,
                long long abatch, long long wbatch, long long cbatch)
{
    __shared__ float As[64][20];    // pad to 20: 16B-aligned rows, no bank conflicts
    __shared__ float Bs[128][20];   // Bs[n][k] == W[tileN+n][k0+k]

    const float* Ab = A + (long long)blockIdx.z * abatch;
    const float* Wb = W + (long long)blockIdx.z * wbatch;
    float*       Cb = Cp + (long long)blockIdx.z * cbatch;

    const int tileM = blockIdx.y * 64;
    const int tileN = blockIdx.x * 128;

    const int tid  = threadIdx.x;
    const int wid  = tid >> 5;
    const int lane = tid & 31;
    const int wm   = wid & 3;      // wave M index 0..3
    const int wn   = wid >> 2;     // wave N index 0..1
    const int l16  = lane & 15;
    const int kh   = lane >> 4;    // 0/1 (half-wave)

    v8f acc[4] = {};

    // A staging: 64 rows x 16 cols = 256 float4, one per thread
    const int ar   = tid >> 2;             // 0..63
    const int ak   = (tid & 3) << 2;       // 0,4,8,12
    const int arow = min(tileM + ar, M - 1);
    // W staging: 128 rows x 16 cols = 512 float4, two per thread
    const int q0   = tid << 1;

    for (int k0 = 0; k0 < K; k0 += 16) {
        {
            const float4 v = *(const float4*)(Ab + (long long)arow * lda + k0 + ak);
            As[ar][ak + 0] = v.x; As[ar][ak + 1] = v.y;
            As[ar][ak + 2] = v.z; As[ar][ak + 3] = v.w;
        }
        #pragma unroll
        for (int u = 0; u < 2; ++u) {
            const int q  = q0 + u;
            const int n  = q >> 2;             // 0..127
            const int kk = (q & 3) << 2;       // 0,4,8,12
            const int wrow = min(tileN + n, N - 1);
            const float4 v = *(const float4*)(Wb + (long long)wrow * ldw + k0 + kk);
            Bs[n][kk + 0] = v.x; Bs[n][kk + 1] = v.y;
            Bs[n][kk + 2] = v.z; Bs[n][kk + 3] = v.w;
        }
        __syncthreads();

        const int lrow = wm * 16 + l16;
        #pragma unroll
        for (int kc = 0; kc < 4; ++kc) {
            // f32 A frag 16x4: v0 holds K=kb (lanes 0-15) / K=kb (lanes 16-31 with kh offset)
            const int kb = kc * 4 + 2 * kh;
            v2f a;
            a.x = As[lrow][kb];
            a.y = As[lrow][kb + 1];
            #pragma unroll
            for (int s = 0; s < 4; ++s) {
                const int lcol = wn * 64 + s * 16 + l16;
                v2f b;
                b.x = Bs[lcol][kb];        // B[k][n] = W[n][k]
                b.y = Bs[lcol][kb + 1];
                acc[s] = __builtin_amdgcn_wmma_f32_16x16x4_f32(
                    false, a, false, b, (short)0, acc[s], false, false);
            }
        }
        __syncthreads();
    }

    // C/D layout: vgpr v -> M = v (lanes 0-15) or v+8 (lanes 16-31); N = lane%16
    #pragma unroll
    for (int s = 0; s < 4; ++s) {
        const int col = tileN + wn * 64 + s * 16 + l16;
        if (col < N) {
            #pragma unroll
            for (int v = 0; v < 8; ++v) {
                const int row = tileM + wm * 16 + v + 8 * kh;
                if (row < M) Cb[(long long)row * ldc + col] = acc[s][v];
            }
        }
    }
}

// ---------------------------------------------------------------------------
// Per-timestep attention recursion (single block).
// d[b,tau,x,y] = q_i - 2*S_ij + q_j  (i=(tau,x), j=(t,y)), clamped [0, 1e5].
// mn over dist (with appended zeros) is exactly 0, so sim = 1 - d / max(d).
// attn[y] = sum_tau exp(-tau/4)/3 * sum_x hist[tau][x] * sim[tau,x,y],
// then softmax(attn/TEMP). hist stores the softmaxed attention per (w,t,b,g).
// ---------------------------------------------------------------------------
__global__ __launch_bounds__(512)
void attn_kernel(const float* __restrict__ S, float* __restrict__ hist,
                 const float* __restrict__ first_att, int t)
{
    __shared__ float d_s[B_ * 15 * G_ * G_];   // 8640 floats max
    __shared__ float wred[16];
    __shared__ float mx_s;
    const int tid = threadIdx.x;

    if (t == 0) {
        if (tid < WK_ * B_) {
            const int w = tid / B_, b = tid % B_;
            const float l0 = first_att[w * G_ + 0] * TEMP_INV;
            const float l1 = first_att[w * G_ + 1] * TEMP_INV;
            const float l2 = first_att[w * G_ + 2] * TEMP_INV;
            const float m  = fmaxf(l0, fmaxf(l1, l2));
            const float e0 = expf(l0 - m), e1 = expf(l1 - m), e2 = expf(l2 - m);
            const float inv = 1.0f / (e0 + e1 + e2);
            float* hp = hist + ((w * T_ + 0) * B_ + b) * G_;
            hp[0] = e0 * inv; hp[1] = e1 * inv; hp[2] = e2 * inv;
        }
        return;
    }

    const int total = B_ * t * G_ * G_;
    float lmax = -1.0f;
    for (int i = tid; i < total; i += blockDim.x) {
        const int y   = i % G_;
        const int x   = (i / G_) % G_;
        const int tau = (i / (G_ * G_)) % t;
        const int b   = i / (G_ * G_ * t);
        const float* Sb = S + b * NR_ * NR_;
        const int ii = tau * G_ + x;
        const int jj = t * G_ + y;
        float d = Sb[ii * NR_ + ii] - 2.0f * Sb[ii * NR_ + jj] + Sb[jj * NR_ + jj];
        d = fminf(fmaxf(d, 0.0f), CLAMP_MAX_);
        d_s[i] = d;
        lmax = fmaxf(lmax, d);
    }
    #pragma unroll
    for (int off = 16; off > 0; off >>= 1)
        lmax = fmaxf(lmax, __shfl_xor(lmax, off, 32));
    if ((tid & 31) == 0) wred[tid >> 5] = lmax;
    __syncthreads();
    if (tid == 0) {
        float m = wred[0];
        for (int i = 1; i < (int)(blockDim.x >> 5); ++i) m = fmaxf(m, wred[i]);
        mx_s = (m > 0.0f) ? m : 1.0f;
    }
    __syncthreads();

    if (tid < WK_ * B_) {
        const int w = tid / B_, b = tid % B_;
        const float inv_mx = 1.0f / mx_s;
        float acc0 = 0.0f, acc1 = 0.0f, acc2 = 0.0f;
        for (int tau = 0; tau < t; ++tau) {
            const float dec = expf(-(float)tau * 0.25f) * (1.0f / 3.0f);
            const float* hp = hist + ((w * T_ + tau) * B_ + b) * G_;
            const float a0 = hp[0], a1 = hp[1], a2 = hp[2];
            const float* dp = d_s + (b * t + tau) * (G_ * G_);
            const float s0 = a0 * (1.0f - dp[0] * inv_mx) + a1 * (1.0f - dp[3] * inv_mx) + a2 * (1.0f - dp[6] * inv_mx);
            const float s1 = a0 * (1.0f - dp[1] * inv_mx) + a1 * (1.0f - dp[4] * inv_mx) + a2 * (1.0f - dp[7] * inv_mx);
            const float s2 = a0 * (1.0f - dp[2] * inv_mx) + a1 * (1.0f - dp[5] * inv_mx) + a2 * (1.0f - dp[8] * inv_mx);
            acc0 += dec * s0; acc1 += dec * s1; acc2 += dec * s2;
        }
        const float l0 = acc0 * TEMP_INV, l1 = acc1 * TEMP_INV, l2 = acc2 * TEMP_INV;
        const float m  = fmaxf(l0, fmaxf(l1, l2));
        const float e0 = expf(l0 - m), e1 = expf(l1 - m), e2 = expf(l2 - m);
        const float inv = 1.0f / (e0 + e1 + e2);
        float* hp = hist + ((w * T_ + t) * B_ + b) * G_;
        hp[0] = e0 * inv; hp[1] = e1 * inv; hp[2] = e2 * inv;
    }
}

// ---------------------------------------------------------------------------
// GRU gates + state update. gi is reconstructed from precomputed P via the
// attention weights (gi = sum_g attn_g * P[w,b,t,g,:] + bih). h_prev is ALWAYS
// list_r[0] (reference bug reproduced). Accumulates hsum for the final FC.
// ---------------------------------------------------------------------------
__global__ __launch_bounds__(256)
void gru_pointwise(const float* __restrict__ gbuf, const float* __restrict__ P,
                   const float* __restrict__ hist,
                   const float* __restrict__ bih, const float* __restrict__ bhh,
                   float* __restrict__ h, float* __restrict__ hsum,
                   int t, int w_base)
{
    const int w   = w_base + blockIdx.z;
    const int idx = blockIdx.x * blockDim.x + threadIdx.x;   // 0..32767
    const int b   = idx >> 9;
    const int j   = idx & 511;

    const float* hp = hist + ((w * T_ + t) * B_ + b) * G_;
    const float a0 = hp[0], a1 = hp[1], a2 = hp[2];
    const float* Pb = P + ((long long)w * (B_ * NR_) + (long long)(b * NR_ + t * G_)) * TH3_;
    const float* gb = gbuf + (long long)(w * B_ + b) * TH3_;
    const float* bihw = bih + w * TH3_;
    const float* bhhw = bhh + w * TH3_;

    float gi[3], gh[3];
    #pragma unroll
    for (int k = 0; k < 3; ++k) {
        const int jj = j + k * H_;
        gi[k] = bihw[jj] + a0 * Pb[jj] + a1 * Pb[TH3_ + jj] + a2 * Pb[2 * TH3_ + jj];
        gh[k] = bhhw[jj] + gb[jj];
    }
    const float hprev = h[b * H_ + j];   // list_r[0] always (reference bug)
    const float r = 1.0f / (1.0f + expf(-(gi[0] + gh[0])));
    const float z = 1.0f / (1.0f + expf(-(gi[1] + gh[1])));
    const float n = tanhf(gi[2] + r * gh[2]);
    const float hn = (1.0f - z) * n + z * hprev;
    h[(w * B_ + b) * H_ + j] = hn;
    hsum[(w * B_ + b) * H_ + j] += hn;
}

// out[b][c] = (1/48) * sum_w ( hsum_w[b] . fc_w[w][c] + 16 * fc_b[w][c] )
__global__ void fc_kernel(const float* __restrict__ hsum, const float* __restrict__ fc_w,
                          const float* __restrict__ fc_b, float* __restrict__ out)
{
    const int c = blockIdx.x;     // 60
    const int b = threadIdx.x;    // 64
    float acc = 0.0f;
    for (int w = 0; w < WK_; ++w) {
        const float4* hs = (const float4*)(hsum + (w * B_ + b) * H_);
        const float4* fw = (const float4*)(fc_w + (w * C_ + c) * H_);
        float dot = 0.0f;
        for (int i = 0; i < H_ / 4; ++i) {
            const float4 hv = hs[i], wv = fw[i];
            dot += hv.x * wv.x + hv.y * wv.y + hv.z * wv.z + hv.w * wv.w;
        }
        acc += dot + 16.0f * fc_b[w * C_ + c];
    }
    out[b * C_ + c] = acc * (1.0f / 48.0f);
}

__global__ void zero_kernel(float* __restrict__ p, int n)
{
    const int i = blockIdx.x * blockDim.x + threadIdx.x;
    if (i < n) p[i] = 0.0f;
}

// ---------------------------------------------------------------------------
extern "C" void kernel_launch(void* const* d_in, const int* in_sizes, int n_in,
                              void* d_out, int out_size, void* d_ws, size_t ws_size,
                              hipStream_t stream)
{
    const float* all_v     = (const float*)d_in[0];  // (B,T,G,D) == (3072, 2048)
    const float* inv_cov   = (const float*)d_in[1];  // (D,D) symmetric
    const float* first_att = (const float*)d_in[2];  // (WK,G)
    const float* gru_wih   = (const float*)d_in[3];  // (WK,3H,D)
    const float* gru_whh   = (const float*)d_in[4];  // (WK,3H,H)
    const float* gru_bih   = (const float*)d_in[5];  // (WK,3H)
    const float* gru_bhh   = (const float*)d_in[6];  // (WK,3H)
    const float* fc_w      = (const float*)d_in[7];  // (WK,C,H)
    const float* fc_b      = (const float*)d_in[8];  // (WK,C)
    float* out = (float*)d_out;

    float* ws   = (float*)d_ws;
    float* U    = ws;                                   // 3072*2048
    float* P    = U + (size_t)3072 * 2048;              // 3*3072*1536
    float* S    = P + (size_t)3 * 3072 * 1536;          // 64*48*48
    float* hist = S + (size_t)64 * 48 * 48;             // 3*16*64*3
    float* h    = hist + (size_t)3 * 16 * 64 * 3;       // 3*64*512
    float* hsum = h + (size_t)3 * 64 * 512;             // 3*64*512 (contiguous with h)
    float* gbuf = hsum + (size_t)3 * 64 * 512;          // 3*64*1536

    // zero h and hsum (contiguous)
    zero_kernel<<<(2 * 3 * 64 * 512 + 255) / 256, 256, 0, stream>>>(h, 2 * 3 * 64 * 512);

    // U = V @ M (M symmetric, so V @ M^T == V @ M)
    gemm_at_wt<<<dim3(2048 / 128, 3072 / 64, 1), 256, 0, stream>>>(
        all_v, inv_cov, U, 3072, 2048, 2048, 2048, 2048, 2048, 0, 0, 0);

    // P[w] = V @ wih[w]^T   (batched over w)
    gemm_at_wt<<<dim3(1536 / 128, 3072 / 64, 3), 256, 0, stream>>>(
        all_v, gru_wih, P, 3072, 1536, 2048, 2048, 2048, 1536,
        0, (long long)1536 * 2048, (long long)3072 * 1536);

    // S[b] = V_b @ U_b^T    (batched over b; 48x48, guarded tile)
    gemm_at_wt<<<dim3(1, 1, 64), 256, 0, stream>>>(
        all_v, U, S, 48, 48, 2048, 2048, 2048, 48,
        (long long)48 * 2048, (long long)48 * 2048, (long long)48 * 48);

    for (int t = 0; t < T_; ++t) {
        attn_kernel<<<1, 512, 0, stream>>>(S, hist, first_att, t);

        // w = 0: gh = h0_{t-1} @ whh[0]^T ; then GRU update of h0 (in place)
        gemm_at_wt<<<dim3(1536 / 128, 1, 1), 256, 0, stream>>>(
            h, gru_whh, gbuf, 64, 1536, 512, 512, 512, 1536, 0, 0, 0);
        gru_pointwise<<<dim3(128, 1, 1), 256, 0, stream>>>(
            gbuf, P, hist, gru_bih, gru_bhh, h, hsum, t, 0);

        // w = 1,2: both read the *updated* h0_t (reference bug), batched via z
        gemm_at_wt<<<dim3(1536 / 128, 1, 2), 256, 0, stream>>>(
            h, gru_whh + (size_t)1536 * 512, gbuf + (size_t)64 * 1536,
            64, 1536, 512, 512, 512, 1536,
            0, (long long)1536 * 512, (long long)64 * 1536);
        gru_pointwise<<<dim3(128, 1, 2), 256, 0, stream>>>(
            gbuf, P, hist, gru_bih, gru_bhh, h, hsum, t, 1);
    }

    fc_kernel<<<60, 64, 0, stream>>>(hsum, fc_w, fc_b, out);
}